// IPA_80530636800064
// MI455X (gfx1250) — compile-verified
//
#include <hip/hip_runtime.h>
#include <math.h>

typedef __attribute__((ext_vector_type(2))) float v2f;
typedef __attribute__((ext_vector_type(8))) float v8f;

#define B_  2
#define N_  2048
#define K_  30
#define D_  128
#define H_  4
#define PQ_ 8
#define PV_ 4
#define BN_ (B_ * N_)
#define CATW 1088   // H*(D + D + 4*PV)

#define INV3C_   0.05103103630798288f   // sqrt(1/(3*128))
#define INV3_    0.5773502691896258f    // sqrt(1/3)
#define PT_SCALE_ 0.09622504486493764f  // sqrt(1/(3*(PQ*9/2)))

// ---------------------------------------------------------------------------
// Generic f32 GEMM: Y[M,N] = A[M,K] @ W[K,N] + bias[N]  (optional ReLU)
// One wave (32 threads) computes one 16x16 tile via v_wmma_f32_16x16x4_f32.
// A frag (16x4): lanes 0-15 -> K=k0+0/k0+1 in a.x/a.y; lanes 16-31 -> K=k0+2/3.
// B frag (4x16): lanes 0-15 -> rows k0+0/k0+1; lanes 16-31 -> rows k0+2/3.
// C/D: VGPR r -> M = r (lanes 0-15) or r+8 (lanes 16-31), N = lane&15.
// ---------------------------------------------------------------------------
__global__ void gemm_wmma_f32_kernel(const float* __restrict__ A,
                                     const float* __restrict__ W,
                                     const float* __restrict__ bias,
                                     float* __restrict__ Y,
                                     int M, int N, int K, int relu)
{
    const int m0   = blockIdx.x * 16;
    const int n0   = blockIdx.y * 16;
    const int lane = threadIdx.x & 31;
    const int half = lane >> 4;      // 0: K pair (0,1); 1: K pair (2,3)
    const int lrow = lane & 15;

    const float* __restrict__ Arow = A + (size_t)(m0 + lrow) * K + half * 2;
    const float* __restrict__ Bcol = W + (size_t)(half * 2) * N + n0 + lrow;

    v8f acc = {};
#pragma unroll 4
    for (int k0 = 0; k0 < K; k0 += 4) {
        v2f a, b;
        a.x = Arow[k0];
        a.y = Arow[k0 + 1];
        b.x = Bcol[(size_t)k0 * N];
        b.y = Bcol[(size_t)(k0 + 1) * N];
        acc = __builtin_amdgcn_wmma_f32_16x16x4_f32(
            /*neg_a=*/false, a, /*neg_b=*/false, b,
            /*c_mod=*/(short)0, acc, /*reuse_a=*/false, /*reuse_b=*/false);
    }

    const int col = n0 + lrow;
    const float bs = bias[col];
#pragma unroll
    for (int r = 0; r < 8; ++r) {
        const int row = m0 + r + half * 8;
        float v = acc[r] + bs;
        if (relu) v = fmaxf(v, 0.0f);
        Y[(size_t)row * N + col] = v;
    }
}

// ---------------------------------------------------------------------------
// b_bias = h_E @ Wb + bb : one thread per edge computes all H=4 heads.
// ---------------------------------------------------------------------------
__global__ void bbias_kernel(const float* __restrict__ hE,
                             const float* __restrict__ Wb,
                             const float* __restrict__ bb,
                             float* __restrict__ bbias)
{
    const int e = blockIdx.x * blockDim.x + threadIdx.x;   // 0 .. B*N*K-1
    if (e >= BN_ * K_) return;
    const float* __restrict__ row = hE + (size_t)e * D_;
    float a0 = bb[0], a1 = bb[1], a2 = bb[2], a3 = bb[3];
    for (int c = 0; c < D_; ++c) {
        const float x = row[c];
        const float* __restrict__ w = Wb + c * H_;
        a0 += x * w[0]; a1 += x * w[1]; a2 += x * w[2]; a3 += x * w[3];
    }
    float* __restrict__ o = bbias + (size_t)e * H_;
    o[0] = a0; o[1] = a1; o[2] = a2; o[3] = a3;
}

// ---------------------------------------------------------------------------
// Rigid transform of projected points: p' = R p + t, reordering the
// split/stack layout (x-major chunks of 32 / 48) into (j,3) layout.
// ---------------------------------------------------------------------------
__global__ void rigid_kernel(const float* __restrict__ qp_raw,   // (BN,96)
                             const float* __restrict__ kvp_raw,  // (BN,144)
                             const float* __restrict__ Rm,       // (BN,9)
                             const float* __restrict__ tv,       // (BN,3)
                             float* __restrict__ qpts,           // (BN,32,3)
                             float* __restrict__ kvpts)          // (BN,48,3)
{
    const int node = blockIdx.x;
    const int tid  = threadIdx.x;
    __shared__ float Rs[9], ts[3];
    if (tid < 9) Rs[tid] = Rm[(size_t)node * 9 + tid];
    if (tid < 3) ts[tid] = tv[(size_t)node * 3 + tid];
    __syncthreads();

    if (tid < 32) {
        const int j = tid;
        const float* __restrict__ src = qp_raw + (size_t)node * 96;
        const float p0 = src[j], p1 = src[32 + j], p2 = src[64 + j];
        float* __restrict__ dst = qpts + (size_t)node * 96 + j * 3;
        dst[0] = Rs[0] * p0 + Rs[1] * p1 + Rs[2] * p2 + ts[0];
        dst[1] = Rs[3] * p0 + Rs[4] * p1 + Rs[5] * p2 + ts[1];
        dst[2] = Rs[6] * p0 + Rs[7] * p1 + Rs[8] * p2 + ts[2];
    } else if (tid < 80) {
        const int j = tid - 32;
        const float* __restrict__ src = kvp_raw + (size_t)node * 144;
        const float p0 = src[j], p1 = src[48 + j], p2 = src[96 + j];
        float* __restrict__ dst = kvpts + (size_t)node * 144 + j * 3;
        dst[0] = Rs[0] * p0 + Rs[1] * p1 + Rs[2] * p2 + ts[0];
        dst[1] = Rs[3] * p0 + Rs[4] * p1 + Rs[5] * p2 + ts[1];
        dst[2] = Rs[6] * p0 + Rs[7] * p1 + Rs[8] * p2 + ts[2];
    }
}

// ---------------------------------------------------------------------------
// Attention core: one 128-thread block per (node, head).
// Phase 1: 30 threads compute logits (q.k dot + point term + bias + mask).
// Phase 2: serial softmax over 30.
// Phase 3: each thread accumulates one channel of o / o_pair; threads 0..11
//          accumulate o_pt components; threads 0..3 rotate back + norm.
// Writes the 1088-wide cat row.
// ---------------------------------------------------------------------------
__global__ void ipa_attn_kernel(const float* __restrict__ q,      // (BN,H,D)
                                const float* __restrict__ kv,     // (BN,H,2D)
                                const float* __restrict__ qpts,   // (BN,H,PQ,3)
                                const float* __restrict__ kvpts,  // (BN,H,PQ+PV,3)
                                const float* __restrict__ bbias,  // (BN,K,H)
                                const float* __restrict__ hE,     // (BN,K,D)
                                const int*   __restrict__ Eidx,   // (BN,K)
                                const float* __restrict__ Rm,     // (BN,9)
                                const float* __restrict__ tv,     // (BN,3)
                                const float* __restrict__ maskA,  // (BN,K)
                                const float* __restrict__ headw,  // (H)
                                float* __restrict__ cat)          // (BN,1088)
{
    const int node = blockIdx.x >> 2;
    const int h    = blockIdx.x & 3;
    const int b    = node / N_;
    const int tid  = threadIdx.x;   // 128

    __shared__ float qs[D_];
    __shared__ float qps[PQ_ * 3];
    __shared__ float Rs[9], ts[3];
    __shared__ float aw[K_];
    __shared__ int   idxs[K_];
    __shared__ float optg[PV_ * 3];

    qs[tid] = q[((size_t)node * H_ + h) * D_ + tid];
    if (tid < PQ_ * 3) qps[tid] = qpts[((size_t)node * H_ + h) * (PQ_ * 3) + tid];
    if (tid < 9) Rs[tid] = Rm[(size_t)node * 9 + tid];
    if (tid < 3) ts[tid] = tv[(size_t)node * 3 + tid];
    __syncthreads();

    if (tid < K_) {
        const int nb = Eidx[(size_t)node * K_ + tid];
        const int nbnode = b * N_ + nb;
        idxs[tid] = nbnode;
        const float* __restrict__ krow = kv + ((size_t)nbnode * H_ + h) * (2 * D_);
        float dot = 0.0f;
        for (int c = 0; c < D_; ++c) dot += qs[c] * krow[c];
        const float* __restrict__ kp =
            kvpts + ((size_t)nbnode * H_ + h) * ((PQ_ + PV_) * 3);
        float pt = 0.0f;
        for (int j = 0; j < PQ_ * 3; ++j) { const float d = qps[j] - kp[j]; pt += d * d; }
        const float hw = log1pf(expf(headw[h]));    // softplus
        float logit = dot * INV3C_
                    + INV3_ * bbias[((size_t)node * K_ + tid) * H_ + h]
                    - 0.5f * PT_SCALE_ * hw * pt
                    + 100000.0f * (maskA[(size_t)node * K_ + tid] - 1.0f);
        aw[tid] = logit;
    }
    __syncthreads();

    if (tid == 0) {
        float mx = -1e30f;
        for (int k = 0; k < K_; ++k) mx = fmaxf(mx, aw[k]);
        float s = 0.0f;
        for (int k = 0; k < K_; ++k) { const float e = expf(aw[k] - mx); aw[k] = e; s += e; }
        const float inv = 1.0f / s;
        for (int k = 0; k < K_; ++k) aw[k] *= inv;
    }
    __syncthreads();

    // o (H*D) and o_pair (H*D)
    float o = 0.0f, op = 0.0f;
    for (int k = 0; k < K_; ++k) {
        const float a = aw[k];
        const size_t nbnode = (size_t)idxs[k];
        o  += a * kv[(nbnode * H_ + h) * (2 * D_) + D_ + tid];
        op += a * hE[((size_t)node * K_ + k) * D_ + tid];
    }
    float* __restrict__ catr = cat + (size_t)node * CATW;
    catr[h * D_ + tid]                       = o;
    catr[2 * H_ * PV_ * 2 + 64 + h * D_ + tid] = op;  // 128 + h*128 + tid

    // o_pt accumulation in global frame (12 components)
    if (tid < PV_ * 3) {
        float g = 0.0f;
        for (int k = 0; k < K_; ++k) {
            const size_t nbnode = (size_t)idxs[k];
            g += aw[k] * kvpts[((nbnode * H_ + h) * (PQ_ + PV_) + PQ_) * 3 + tid];
        }
        optg[tid] = g;
    }
    __syncthreads();

    // rotate back: out[x] = sum_y R[y][x] * (g[y] - t[y]); plus norm
    if (tid < PV_) {
        const int p = tid;
        const float gx = optg[p * 3 + 0] - ts[0];
        const float gy = optg[p * 3 + 1] - ts[1];
        const float gz = optg[p * 3 + 2] - ts[2];
        const float lx = Rs[0] * gx + Rs[3] * gy + Rs[6] * gz;
        const float ly = Rs[1] * gx + Rs[4] * gy + Rs[7] * gz;
        const float lz = Rs[2] * gx + Rs[5] * gy + Rs[8] * gz;
        const float nrm = sqrtf(lx * lx + ly * ly + lz * lz + 1e-8f);
        catr[512 +  0 + h * PV_ + p] = lx;
        catr[512 + 16 + h * PV_ + p] = ly;
        catr[512 + 32 + h * PV_ + p] = lz;
        catr[512 + 48 + h * PV_ + p] = nrm;
    }
}

// ---------------------------------------------------------------------------
// LayerNorm over residual sum: out = LN(X + Y) * w + b   (128 threads/row)
// Optional per-row mask multiply (for the final output).
// ---------------------------------------------------------------------------
__global__ void ln_kernel(const float* __restrict__ X,
                          const float* __restrict__ Yd,
                          const float* __restrict__ w,
                          const float* __restrict__ bp,
                          const float* __restrict__ maskV,  // nullptr-like: use_mask==0
                          float* __restrict__ out, int use_mask)
{
    const int row = blockIdx.x;
    const int tid = threadIdx.x;   // 128
    __shared__ float red[D_];

    const float x = X[(size_t)row * D_ + tid] + Yd[(size_t)row * D_ + tid];
    red[tid] = x;
    __syncthreads();
    for (int s = 64; s > 0; s >>= 1) { if (tid < s) red[tid] += red[tid + s]; __syncthreads(); }
    const float mu = red[0] * (1.0f / D_);
    __syncthreads();
    const float d = x - mu;
    red[tid] = d * d;
    __syncthreads();
    for (int s = 64; s > 0; s >>= 1) { if (tid < s) red[tid] += red[tid + s]; __syncthreads(); }
    const float var = red[0] * (1.0f / D_);
    float v = d * rsqrtf(var + 1e-5f) * w[tid] + bp[tid];
    if (use_mask) v *= maskV[row];
    out[(size_t)row * D_ + tid] = v;
}

// ---------------------------------------------------------------------------
extern "C" void kernel_launch(void* const* d_in, const int* in_sizes, int n_in,
                              void* d_out, int out_size, void* d_ws, size_t ws_size,
                              hipStream_t stream)
{
    const float* h_V   = (const float*)d_in[0];
    const float* h_E   = (const float*)d_in[1];
    const int*   E_idx = (const int*)  d_in[2];
    const float* R     = (const float*)d_in[3];
    const float* t     = (const float*)d_in[4];
    const float* maskA = (const float*)d_in[5];
    const float* maskV = (const float*)d_in[6];
    const float* Wq    = (const float*)d_in[7];  const float* bq   = (const float*)d_in[8];
    const float* Wkv   = (const float*)d_in[9];  const float* bkv  = (const float*)d_in[10];
    const float* Wqp   = (const float*)d_in[11]; const float* bqp  = (const float*)d_in[12];
    const float* Wkvp  = (const float*)d_in[13]; const float* bkvp = (const float*)d_in[14];
    const float* Wb    = (const float*)d_in[15]; const float* bb   = (const float*)d_in[16];
    const float* headw = (const float*)d_in[17];
    const float* Wout  = (const float*)d_in[18]; const float* bout = (const float*)d_in[19];
    const float* ln0w  = (const float*)d_in[20]; const float* ln0b = (const float*)d_in[21];
    const float* ln1w  = (const float*)d_in[22]; const float* ln1b = (const float*)d_in[23];
    const float* Wm1   = (const float*)d_in[24]; const float* bm1  = (const float*)d_in[25];
    const float* Wm2   = (const float*)d_in[26]; const float* bm2  = (const float*)d_in[27];
    float* out = (float*)d_out;

    // Workspace carve-up (floats)
    float* ws = (float*)d_ws;
    size_t off = 0;
    float* qb      = ws + off; off += (size_t)BN_ * H_ * D_;        // 2.10M
    float* kvb     = ws + off; off += (size_t)BN_ * H_ * 2 * D_;    // 4.19M
    float* qp_raw  = ws + off; off += (size_t)BN_ * 96;
    float* kvp_raw = ws + off; off += (size_t)BN_ * 144;
    float* qpts    = ws + off; off += (size_t)BN_ * 96;
    float* kvpts   = ws + off; off += (size_t)BN_ * 144;
    float* bbias   = ws + off; off += (size_t)BN_ * K_ * H_;
    float* catb    = ws + off; off += (size_t)BN_ * CATW;
    float* sbuf    = ws + off; off += (size_t)BN_ * D_;
    float* hvb     = ws + off; off += (size_t)BN_ * D_;
    float* m1b     = ws + off; off += (size_t)BN_ * 4 * D_;
    float* m2b     = ws + off; off += (size_t)BN_ * D_;
    (void)off; (void)in_sizes; (void)n_in; (void)out_size; (void)ws_size;

    auto gemm = [&](const float* A, const float* W, const float* bias, float* Y,
                    int M, int N, int K, int relu) {
        dim3 grid(M / 16, N / 16);
        gemm_wmma_f32_kernel<<<grid, dim3(32), 0, stream>>>(A, W, bias, Y, M, N, K, relu);
    };

    // 1) Node projections (WMMA f32 GEMMs)
    gemm(h_V, Wq,   bq,   qb,      BN_,  H_ * D_,     D_, 0);   // 4096x512
    gemm(h_V, Wkv,  bkv,  kvb,     BN_,  H_ * 2 * D_, D_, 0);   // 4096x1024
    gemm(h_V, Wqp,  bqp,  qp_raw,  BN_,  96,          D_, 0);   // 4096x96
    gemm(h_V, Wkvp, bkvp, kvp_raw, BN_,  144,         D_, 0);   // 4096x144

    // 2) Edge bias (N=4, plain VALU)
    {
        const int total = BN_ * K_;
        bbias_kernel<<<(total + 255) / 256, 256, 0, stream>>>(h_E, Wb, bb, bbias);
    }

    // 3) Rigid point transforms
    rigid_kernel<<<BN_, 128, 0, stream>>>(qp_raw, kvp_raw, R, t, qpts, kvpts);

    // 4) Attention (gather + softmax + weighted sums -> cat)
    ipa_attn_kernel<<<BN_ * H_, 128, 0, stream>>>(qb, kvb, qpts, kvpts, bbias,
                                                  h_E, E_idx, R, t, maskA, headw, catb);

    // 5) Output projection + LN0
    gemm(catb, Wout, bout, sbuf, BN_, D_, CATW, 0);             // 4096x128, K=1088
    ln_kernel<<<BN_, D_, 0, stream>>>(h_V, sbuf, ln0w, ln0b, maskV, hvb, 0);

    // 6) MLP + LN1 (+ mask) -> out
    gemm(hvb, Wm1, bm1, m1b, BN_, 4 * D_, D_, 1);               // ReLU fused
    gemm(m1b, Wm2, bm2, m2b, BN_, D_, 4 * D_, 0);
    ln_kernel<<<BN_, D_, 0, stream>>>(hvb, m2b, ln1w, ln1b, maskV, out, 1);
}